// TriangleConv_28406913696569
// MI455X (gfx1250) — compile-verified
//
#include <hip/hip_runtime.h>

// ---------------------------------------------------------------------------
// TriangleConv for MI455X (gfx1250), compile-only tuning.
//
// Strategy: X (12.6 MB) lives in L2; recompute GEMMs instead of materializing
// 268 MB intermediates. 3 passes: stats1 -> finalize1 -> stats2 -> finalize2
// -> output. Total HBM traffic ~306 MB (vs ~1.35 GB naive) -> ~13 us at
// 23.3 TB/s. Matrix work uses V_WMMA_F32_16X16X4_F32 (f32-exact).
// Conv biases cancel in training-mode BatchNorm and are dropped.
//
// Round-1: removed lane-varying private-array indexing (v_cndmask trees);
//          stats accumulate in registers, LDS atomics once per kernel.
// Round-2: LDS operands for GEMM2 loaded as aligned v2f -> ds_load_b64;
//          final output stored non-temporal (write-once stream, keeps X/W
//          resident in L2 across passes).
// ---------------------------------------------------------------------------

typedef float v2f __attribute__((ext_vector_type(2), aligned(8)));
typedef float v8f __attribute__((ext_vector_type(8)));

#define WMMA_F32X4(a, b, c) \
  __builtin_amdgcn_wmma_f32_16x16x4_f32(false, (a), false, (b), (short)0, (c), false, false)

namespace {
constexpr int kBatch  = 16;
constexpr int kN      = 65536;       // points per batch (power of two -> cheap ring wrap)
constexpr int kF      = 3;           // raw features
constexpr int kK1     = 12;          // 4*F input channels of layer 1
constexpr int kC      = 64;          // channels of layer 1/2 outputs
constexpr int kPoints = kBatch * kN; // 1,048,576
constexpr int kTiles  = kPoints / 16;
constexpr int kWaves  = 8;           // waves per block (256 threads, wave32)
constexpr int kTPW    = 8;           // 16-point tiles per wave
constexpr int kBlocks = kTiles / (kWaves * kTPW); // 1024
constexpr float kEps  = 1e-5f;

// Workspace layout (floats):
//  [0,64)    sum1    [64,128)  sumsq1
//  [128,192) sum2    [192,256) sumsq2
//  [256,320) A1      [320,384) C1      (BN1 folded affine: relu(A*y+C))
//  [384,448) A2      [448,512) C2
} // namespace

__global__ void zero_stats_kernel(float* __restrict__ ws) {
  if (threadIdx.x < 256) ws[threadIdx.x] = 0.0f;
}

__global__ void finalize_kernel(const float* __restrict__ g,
                                const float* __restrict__ beta,
                                float* __restrict__ ws, int stage) {
  int c = threadIdx.x;
  if (c >= kC) return;
  const float inv = 1.0f / (float)kPoints;
  const float* s = ws + stage * 128;
  float mean = s[c] * inv;
  float var  = s[c + kC] * inv - mean * mean;   // biased variance (torch BN default)
  float rstd = rsqrtf(var + kEps);
  float a = g[c] * rstd;
  ws[256 + stage * 128 + c]      = a;
  ws[256 + stage * 128 + kC + c] = beta[c] - mean * a;
}

// MODE 0: stats of y1'   MODE 1: stats of y2'   MODE 2: write final output
template <int MODE>
__global__ __launch_bounds__(256) void pass_kernel(
    const float* __restrict__ X, const float* __restrict__ W1,
    const float* __restrict__ W2, float* __restrict__ ws,
    float* __restrict__ out) {
  __shared__ float sW2[kC * kC];            // 16 KB: layer-2 weights
  __shared__ float sH1[kWaves * 16 * kC];   // 32 KB: per-wave C/D->A transpose buffer
  __shared__ float sSum[kC];
  __shared__ float sSq[kC];

  const int  tid   = threadIdx.x;
  const int  lane  = tid & 31;
  const int  wave  = tid >> 5;
  const int  m     = lane & 15;             // row/col index within 16-wide tile
  const bool hi    = (lane >= 16);          // upper half-wave (ISA operand layout)
  const int  koff  = hi ? 2 : 0;            // only used in *address* arithmetic
  const int  mbase = hi ? 8 : 0;            // C/D row base (ISA layout)

  if (MODE < 2 && tid < kC) { sSum[tid] = 0.0f; sSq[tid] = 0.0f; }
  if (MODE >= 1) {
    for (int i = tid; i < kC * kC; i += 256) sW2[i] = W2[i];
  }
  __syncthreads();

  // Layer-1 B operands (W1^T tiles) hoisted into registers: 4 N-tiles x 3 k-steps.
  v2f bw1[4][3];
#pragma unroll
  for (int t = 0; t < 4; ++t)
#pragma unroll
    for (int ks = 0; ks < 3; ++ks) {
      const float* w = W1 + (16 * t + m) * kK1 + 4 * ks + koff;  // address math only
      v2f b; b.x = w[0]; b.y = w[1];
      bw1[t][ks] = b;
    }

  // Folded BN affines for this lane's 4 channels (16*t + m).
  float a1v[4], c1v[4], a2v[4], c2v[4];
  if (MODE >= 1) {
#pragma unroll
    for (int t = 0; t < 4; ++t) {
      a1v[t] = ws[256 + 16 * t + m];
      c1v[t] = ws[320 + 16 * t + m];
    }
  }
  if (MODE == 2) {
#pragma unroll
    for (int t = 0; t < 4; ++t) {
      a2v[t] = ws[384 + 16 * t + m];
      c2v[t] = ws[448 + 16 * t + m];
    }
  }

  // Per-lane stats accumulators (channel 16*t + m), summed across all tiles.
  float accS[4] = {0.f, 0.f, 0.f, 0.f};
  float accQ[4] = {0.f, 0.f, 0.f, 0.f};

  float* hw = &sH1[wave * 16 * kC];
  const int tile0 = (blockIdx.x * kWaves + wave) * kTPW;

  for (int it = 0; it < kTPW; ++it) {
    const int tile = tile0 + it;
    const int p0   = tile * 16;             // tile never straddles a batch row
    const int b    = p0 >> 16;              // / kN
    const int n    = (p0 & (kN - 1)) + m;
    const int np   = (n - 1) & (kN - 1);    // ring wrap (roll +1)
    const int nx   = (n + 1) & (kN - 1);    // ring wrap (roll -1)
    const float* Xb = X + (size_t)b * kN * kF;

    // Triangle features: [x, x-xp, x-xn, xn-xp]  (12 channels)
    float f[12];
#pragma unroll
    for (int j = 0; j < 3; ++j) {
      float x  = Xb[n  * 3 + j];
      float xp = Xb[np * 3 + j];
      float xn = Xb[nx * 3 + j];
      f[j]     = x;
      f[3 + j] = x - xp;
      f[6 + j] = x - xn;
      f[9 + j] = xn - xp;
    }

    // A operands with compile-time indices; one boolean select per component
    // (lanes 0-15 carry K = 4ks+0,1; lanes 16-31 carry K = 4ks+2,3).
    v2f aop[3];
#pragma unroll
    for (int ks = 0; ks < 3; ++ks) {
      aop[ks].x = hi ? f[4 * ks + 2] : f[4 * ks + 0];
      aop[ks].y = hi ? f[4 * ks + 3] : f[4 * ks + 1];
    }

    // -------- GEMM1: [16 x 12] x [12 x 64] via 3 k-steps x 4 N-tiles --------
    v8f d1[4] = {};
#pragma unroll
    for (int ks = 0; ks < 3; ++ks)
#pragma unroll
      for (int t = 0; t < 4; ++t) d1[t] = WMMA_F32X4(aop[ks], bw1[t][ks], d1[t]);

    if (MODE == 0) {
#pragma unroll
      for (int t = 0; t < 4; ++t)
#pragma unroll
        for (int r = 0; r < 8; ++r) {
          float v = d1[t][r];
          accS[t] += v;
          accQ[t] = fmaf(v, v, accQ[t]);
        }
    } else {
      // BN1 + ReLU, then C/D-layout -> A-layout transpose through LDS.
#pragma unroll
      for (int t = 0; t < 4; ++t)
#pragma unroll
        for (int r = 0; r < 8; ++r) {
          float v = fmaxf(fmaf(a1v[t], d1[t][r], c1v[t]), 0.0f);
          hw[(mbase + r) * kC + 16 * t + m] = v;
        }
      // Same-wave LDS RAW: hardware DScnt ordering, no barrier required.

      // -------- GEMM2: [16 x 64] x [64 x 64] via 16 k-steps x 4 N-tiles ------
      // Operands are 8B-aligned adjacent pairs: load as v2f -> ds_load_b64.
      const v2f* hrow = (const v2f*)(hw + m * kC + koff);           // + 4*ks*4B
      v8f d2[4] = {};
#pragma unroll
      for (int ks = 0; ks < 16; ++ks) {
        v2f a = hrow[2 * ks];                                       // (4*ks+koff)*4B
#pragma unroll
        for (int t = 0; t < 4; ++t) {
          const v2f* wrow = (const v2f*)(sW2 + (16 * t + m) * kC + koff);
          d2[t] = WMMA_F32X4(a, wrow[2 * ks], d2[t]);
        }
      }

      if (MODE == 1) {
#pragma unroll
        for (int t = 0; t < 4; ++t)
#pragma unroll
          for (int r = 0; r < 8; ++r) {
            float v = d2[t][r];
            accS[t] += v;
            accQ[t] = fmaf(v, v, accQ[t]);
          }
      } else {
        // BN2 + ReLU + non-temporal store of the final [point, 64] tile
        // (write-once stream; keep X/W resident in L2 for later passes).
#pragma unroll
        for (int t = 0; t < 4; ++t)
#pragma unroll
          for (int r = 0; r < 8; ++r) {
            float v = fmaxf(fmaf(a2v[t], d2[t][r], c2v[t]), 0.0f);
            __builtin_nontemporal_store(
                v, out + (size_t)(p0 + mbase + r) * kC + 16 * t + m);
          }
      }
    }
  }

  if (MODE < 2) {
    // One LDS combine per lane (not per tile), then one global combine per block.
#pragma unroll
    for (int t = 0; t < 4; ++t) {
      atomicAdd(&sSum[16 * t + m], accS[t]);
      atomicAdd(&sSq [16 * t + m], accQ[t]);
    }
    __syncthreads();
    if (tid < kC) {
      float* gs = ws + (MODE == 0 ? 0 : 128);
      atomicAdd(&gs[tid],      sSum[tid]);
      atomicAdd(&gs[tid + kC], sSq[tid]);
    }
  }
}

extern "C" void kernel_launch(void* const* d_in, const int* in_sizes, int n_in,
                              void* d_out, int out_size, void* d_ws, size_t ws_size,
                              hipStream_t stream) {
  (void)in_sizes; (void)n_in; (void)out_size; (void)ws_size;
  const float* X     = (const float*)d_in[0];
  const float* W1    = (const float*)d_in[1];
  // d_in[2] = b1: cancels exactly in training-mode BatchNorm -> unused.
  const float* g1    = (const float*)d_in[3];
  const float* beta1 = (const float*)d_in[4];
  const float* W2    = (const float*)d_in[5];
  // d_in[6] = b2: unused for the same reason.
  const float* g2    = (const float*)d_in[7];
  const float* beta2 = (const float*)d_in[8];
  float* ws  = (float*)d_ws;
  float* out = (float*)d_out;

  zero_stats_kernel<<<1, 256, 0, stream>>>(ws);
  pass_kernel<0><<<kBlocks, 256, 0, stream>>>(X, W1, W2, ws, out);   // stats of y1'
  finalize_kernel<<<1, 64, 0, stream>>>(g1, beta1, ws, 0);           // -> A1, C1
  pass_kernel<1><<<kBlocks, 256, 0, stream>>>(X, W1, W2, ws, out);   // stats of y2'
  finalize_kernel<<<1, 64, 0, stream>>>(g2, beta2, ws, 1);           // -> A2, C2
  pass_kernel<2><<<kBlocks, 256, 0, stream>>>(X, W1, W2, ws, out);   // final write
}